// BatchHardTripletLoss_39779987096332
// MI455X (gfx1250) — compile-verified
//
#include <hip/hip_runtime.h>
#include <hip/hip_bf16.h>
#include <math.h>
#include <float.h>

typedef float v2f __attribute__((ext_vector_type(2)));
typedef float v8f __attribute__((ext_vector_type(8)));

#define EPSF       1e-12f
#define B_N        4096
#define D_K        128
#define LDS_STRIDE 132              // 128 + 4 pad: keeps 16B alignment, kills bank conflicts
#define LDS_TILE   (16 * LDS_STRIDE)
#define LDS_BUF    (3 * LDS_TILE)   // one double-buffer half: 3 source tiles

// ---------------- Kernel 0: row norms + init ----------------
__global__ __launch_bounds__(256) void bh_init(const float* __restrict__ ra,
                                               const float* __restrict__ rp,
                                               const float* __restrict__ rn,
                                               float* __restrict__ norms,
                                               float* __restrict__ minsq,
                                               float* __restrict__ out) {
  int i = blockIdx.x * 256 + threadIdx.x;
  if (i == 0) out[0] = 0.0f;
  if (i >= B_N) return;
  const float4* a4 = (const float4*)(ra + (size_t)i * D_K);
  const float4* p4 = (const float4*)(rp + (size_t)i * D_K);
  const float4* n4 = (const float4*)(rn + (size_t)i * D_K);
  float sa = 0.f, sp = 0.f, sn = 0.f;
#pragma unroll
  for (int t = 0; t < D_K / 4; ++t) {
    float4 a = a4[t], p = p4[t], n = n4[t];
    sa += a.x * a.x + a.y * a.y + a.z * a.z + a.w * a.w;
    sp += p.x * p.x + p.y * p.y + p.z * p.z + p.w * p.w;
    sn += n.x * n.x + n.y * n.y + n.z * n.z + n.w * n.w;
  }
  norms[i]           = sa;
  norms[B_N + i]     = sp;
  norms[2 * B_N + i] = sn;
  minsq[i]           = FLT_MAX;
}

// Issue this thread's share (6 x B128) of one j-tile's staging as CDNA5 async
// global->LDS copies.  VDST operand = LDS byte offset (low 32 bits of a generic
// pointer to __shared__ per the gfx1250 aperture model); tracked on ASYNCcnt.
__device__ __forceinline__ void stage_async(const float* __restrict__ ra,
                                            const float* __restrict__ rp,
                                            const float* __restrict__ rn,
                                            float* ldsbase, int tid, int j0) {
#pragma unroll
  for (int q = 0; q < 6; ++q) {
    const int u   = q * 256 + tid;        // 0..1535
    const int src = u >> 9;               // 0=anchor 1=pos 2=neg
    const int rem = u & 511;
    const int row = rem >> 5;             // 0..15
    const int c4  = rem & 31;             // 0..31 (float4 column)
    const float* base = (src == 0) ? ra : (src == 1) ? rp : rn;
    const float* gp = base + (size_t)(j0 + row) * D_K + c4 * 4;
    float*       lp = ldsbase + src * LDS_TILE + row * LDS_STRIDE + c4 * 4;
    const unsigned           loff = (unsigned)(size_t)lp;       // LDS byte offset
    const unsigned long long ga   = (unsigned long long)gp;     // 64-bit global addr
    asm volatile("global_load_async_to_lds_b128 %0, %1, off"
                 :: "v"(loff), "v"(ga) : "memory");
  }
}

__device__ __forceinline__ void wait_async_all() {
  asm volatile("s_wait_asynccnt 0" ::: "memory");
}

// ---------------- Kernel 1: WMMA Gram distances + row-min ----------------
// 8 waves per WG; wave w owns i-tile (blockIdx.x*8 + w): 16 anchor rows, with the
// full 16x128 A tile resident in 64 VGPRs.  blockIdx.y selects a 32-j-tile chunk.
// B tiles (anchor/pos/neg) are double-buffered in LDS via async copies so global
// latency overlaps the WMMA stream.
__global__ __launch_bounds__(256) void bh_main(const float* __restrict__ ra,
                                               const float* __restrict__ rp,
                                               const float* __restrict__ rn,
                                               const float* __restrict__ norms,
                                               float* __restrict__ minsq) {
  __shared__ float lds[2 * LDS_BUF];            // ~50 KB of the 320 KB WGP pool

  const int tid  = threadIdx.x;
  const int wave = tid >> 5;
  const int lane = tid & 31;
  const int h    = lane >> 4;                   // lane half: K pair / M block select
  const int l16  = lane & 15;
  const int i0   = (blockIdx.x * 8 + wave) * 16;

  const float* na = norms;
  const float* np = norms + B_N;
  const float* nn = norms + 2 * B_N;

  // A fragments for all K=128: lane(l16,h) holds row i0+l16, K = 4t + 2h + {0,1}
  v2f afrag[32];
  {
    const float* arow = ra + (size_t)(i0 + l16) * D_K + 2 * h;
#pragma unroll
    for (int t = 0; t < 32; ++t) afrag[t] = *(const v2f*)(arow + 4 * t);
  }

  float x2row[8];
#pragma unroll
  for (int r = 0; r < 8; ++r) x2row[r] = na[i0 + 8 * h + r];

  float rmin[8];
#pragma unroll
  for (int r = 0; r < 8; ++r) rmin[r] = FLT_MAX;

  const int jt0 = blockIdx.y * 32;

  // Prologue: stage tile 0 into buffer 0.
  stage_async(ra, rp, rn, &lds[0], tid, jt0 * 16);
  wait_async_all();
  __syncthreads();

  for (int jt = 0; jt < 32; ++jt) {
    const int b  = jt & 1;
    const int j0 = (jt0 + jt) * 16;

    // Prefetch next tile into the other buffer; latency hides under the WMMAs.
    if (jt + 1 < 32)
      stage_async(ra, rp, rn, &lds[(b ^ 1) * LDS_BUF], tid, (jt0 + jt + 1) * 16);

    v8f caa = {}, cap = {}, can = {};
    const float* lA = &lds[b * LDS_BUF + 0 * LDS_TILE + l16 * LDS_STRIDE + 2 * h];
    const float* lP = &lds[b * LDS_BUF + 1 * LDS_TILE + l16 * LDS_STRIDE + 2 * h];
    const float* lN = &lds[b * LDS_BUF + 2 * LDS_TILE + l16 * LDS_STRIDE + 2 * h];
#pragma unroll
    for (int t = 0; t < 32; ++t) {
      v2f ba = *(const v2f*)(lA + 4 * t);
      v2f bp = *(const v2f*)(lP + 4 * t);
      v2f bn = *(const v2f*)(lN + 4 * t);
      caa = __builtin_amdgcn_wmma_f32_16x16x4_f32(false, afrag[t], false, ba,
                                                  (short)0, caa, false, false);
      cap = __builtin_amdgcn_wmma_f32_16x16x4_f32(false, afrag[t], false, bp,
                                                  (short)0, cap, false, false);
      can = __builtin_amdgcn_wmma_f32_16x16x4_f32(false, afrag[t], false, bn,
                                                  (short)0, can, false, false);
    }

    // Epilogue: dot -> squared distance, diagonal mask, running min.
    const int   jg  = j0 + l16;                 // this lane's column (N = l16)
    const float y2a = na[jg], y2p = np[jg], y2n = nn[jg];
#pragma unroll
    for (int r = 0; r < 8; ++r) {
      const int   gi = i0 + 8 * h + r;          // C layout: M = r + 8h
      const float x2 = x2row[r];
      float saa = fmaxf(x2 + y2a - 2.0f * caa[r], 0.0f);
      float sap = fmaxf(x2 + y2p - 2.0f * cap[r], 0.0f);
      if (gi == jg) { saa = FLT_MAX; sap = FLT_MAX; }   // exclude self / own positive
      const float san = fmaxf(x2 + y2n - 2.0f * can[r], 0.0f);
      rmin[r] = fminf(rmin[r], fminf(fminf(saa, sap), san));
    }

    // Next tile's async copies landed for THIS wave; barrier makes them visible
    // to all waves and protects buffer b from being restaged while still in use.
    wait_async_all();
    __syncthreads();
  }

  // Min across the 16 lanes of each half (N dimension), then one atomic per row.
#pragma unroll
  for (int r = 0; r < 8; ++r) {
    float m = rmin[r];
    m = fminf(m, __shfl_xor(m, 8, 32));
    m = fminf(m, __shfl_xor(m, 4, 32));
    m = fminf(m, __shfl_xor(m, 2, 32));
    m = fminf(m, __shfl_xor(m, 1, 32));
    if (l16 == 0)  // nonneg floats: int-bit order == float order
      atomicMin(((int*)minsq) + (i0 + 8 * h + r), __float_as_int(m));
  }
}

// ---------------- Kernel 2: dpos + softplus + mean ----------------
__global__ __launch_bounds__(256) void bh_final(const float* __restrict__ ra,
                                                const float* __restrict__ rp,
                                                const float* __restrict__ minsq,
                                                float* __restrict__ out) {
  __shared__ float red[256];
  const int i = blockIdx.x * 256 + threadIdx.x;
  const float4* a4 = (const float4*)(ra + (size_t)i * D_K);
  const float4* p4 = (const float4*)(rp + (size_t)i * D_K);
  float s = 0.f;
#pragma unroll
  for (int t = 0; t < D_K / 4; ++t) {
    float4 a = a4[t], p = p4[t];
    float dx = a.x - p.x, dy = a.y - p.y, dz = a.z - p.z, dw = a.w - p.w;
    s += dx * dx + dy * dy + dz * dz + dw * dw;
  }
  const float dpos = sqrtf(fmaxf(s, EPSF));
  const float hard = sqrtf(fmaxf(minsq[i], EPSF));
  const float x    = dpos - hard;
  const float sp   = fmaxf(x, 0.0f) + log1pf(expf(-fabsf(x)));  // stable logaddexp(0,x)

  red[threadIdx.x] = sp;
  __syncthreads();
  for (int off = 128; off > 0; off >>= 1) {
    if (threadIdx.x < off) red[threadIdx.x] += red[threadIdx.x + off];
    __syncthreads();
  }
  if (threadIdx.x == 0) atomicAdd(out, red[0] * (1.0f / (float)B_N));
}

// ---------------- Launcher ----------------
extern "C" void kernel_launch(void* const* d_in, const int* in_sizes, int n_in,
                              void* d_out, int out_size, void* d_ws, size_t ws_size,
                              hipStream_t stream) {
  const float* ra = (const float*)d_in[0];
  const float* rp = (const float*)d_in[1];
  const float* rn = (const float*)d_in[2];
  float* ws    = (float*)d_ws;
  float* norms = ws;                 // 3*4096 floats
  float* minsq = ws + 3 * B_N;       // 4096 floats
  float* out   = (float*)d_out;

  bh_init <<<dim3(B_N / 256), dim3(256), 0, stream>>>(ra, rp, rn, norms, minsq, out);
  bh_main <<<dim3(32, 8),      dim3(256), 0, stream>>>(ra, rp, rn, norms, minsq);
  bh_final<<<dim3(B_N / 256), dim3(256), 0, stream>>>(ra, rp, minsq, out);
}